// EconomicViewSubEncoder_62989990363720
// MI455X (gfx1250) — compile-verified
//
#include <hip/hip_runtime.h>
#include <math.h>
#include <stdint.h>

#define R_ 25
#define N_ 256
#define C_ 64
#define E_ 128
#define H_ 8
#define NC_ 27

typedef __attribute__((ext_vector_type(2))) float v2f;
typedef __attribute__((ext_vector_type(8))) float v8f;
typedef __attribute__((ext_vector_type(4))) unsigned int u32x4;
typedef __attribute__((ext_vector_type(4))) int i32x4;
typedef __attribute__((ext_vector_type(8))) int i32x8;

#if __has_builtin(__builtin_amdgcn_tensor_load_to_lds) && \
    __has_builtin(__builtin_amdgcn_s_wait_tensorcnt)
#define USE_TDM 1
#else
#define USE_TDM 0
#endif

// D = A(16x4) * B(4x16) + C, fp32 WMMA on gfx1250.
// A per-lane: a[j] = A[M = lane&15][K = 2*(lane>>4) + j]
// B per-lane (mirror layout): b[j] = B[K = 2*(lane>>4)+j][N = lane&15]
// C/D per-lane: c[i] = C[M = i + 8*(lane>>4)][N = lane&15]
__device__ __forceinline__ v8f wmma_f32_4(v2f a, v2f b, v8f c) {
  return __builtin_amdgcn_wmma_f32_16x16x4_f32(false, a, false, b, (short)0, c,
                                               false, false);
}

// ---------------- Kernel 1: layernorm of embedding table (6400 x 128) -------
__global__ void k_emb_ln(const float* __restrict__ emb,
                         const float* __restrict__ w,
                         const float* __restrict__ b,
                         float* __restrict__ out) {
  __shared__ float red[E_];
  int row = blockIdx.x;
  int t = threadIdx.x;  // 128 threads
  float x = emb[(size_t)row * E_ + t];
  red[t] = x;
  __syncthreads();
  for (int s = 64; s > 0; s >>= 1) {
    if (t < s) red[t] += red[t + s];
    __syncthreads();
  }
  float mean = red[0] / (float)E_;
  __syncthreads();
  float d = x - mean;
  red[t] = d * d;
  __syncthreads();
  for (int s = 64; s > 0; s >>= 1) {
    if (t < s) red[t] += red[t + s];
    __syncthreads();
  }
  float var = red[0] / (float)E_;
  out[(size_t)row * E_ + t] = d * rsqrtf(var + 1e-16f) * w[t] + b[t];
}

// ------------- Kernel 2a: mean over trailing 4 (business structure) ---------
__global__ void k_bs_mean(const float* __restrict__ src, float* __restrict__ dst,
                          int rowStride) {
  int row = blockIdx.x, n = threadIdx.x;
  const float* p = src + (size_t)row * rowStride + (size_t)n * 4;
  dst[(size_t)row * N_ + n] = 0.25f * (p[0] + p[1] + p[2] + p[3]);
}

// ---- Kernel 2b: mean over 4 then log_softmax over NC (customer structure) --
__global__ void k_cs_logsoftmax(const float* __restrict__ src,
                                float* __restrict__ dst, int rowStride) {
  int row = blockIdx.x, n = threadIdx.x;
  const float* p = src + (size_t)row * rowStride + (size_t)n * NC_ * 4;
  float v[NC_];
  float mx = -3.4e38f;
  for (int k = 0; k < NC_; ++k) {
    const float* q = p + k * 4;
    float m = 0.25f * (q[0] + q[1] + q[2] + q[3]);
    v[k] = m;
    mx = fmaxf(mx, m);
  }
  float se = 0.f;
  for (int k = 0; k < NC_; ++k) se += expf(v[k] - mx);
  float lse = logf(se);
  float* o = dst + ((size_t)row * N_ + n) * NC_;
  for (int k = 0; k < NC_; ++k) o[k] = v[k] - mx - lse;
}

// -------- Kernel 3: M[s,h] = WQ^T @ WK  (128x128 each, 16 matrices) ---------
__global__ void __launch_bounds__(256) k_wqk(const float* __restrict__ WQt,
                                             const float* __restrict__ WKt,
                                             const float* __restrict__ WQi,
                                             const float* __restrict__ WKi,
                                             float* __restrict__ M) {
  int s = blockIdx.x >> 3, h = blockIdx.x & 7;
  const float* Wq = (s ? WQi : WQt) + (size_t)h * E_ * E_;
  const float* Wk = (s ? WKi : WKt) + (size_t)h * E_ * E_;
  float* Mo = M + (size_t)blockIdx.x * E_ * E_;
  int wave = threadIdx.x >> 5, lane = threadIdx.x & 31;
  int lo = lane & 15, hi = lane >> 4;
  int rt = wave;  // 8 row-tiles, one per wave
  for (int ct = 0; ct < 8; ++ct) {
    v8f acc = {};
    for (int k0 = 0; k0 < E_; k0 += 4) {
      int k = k0 + 2 * hi;
      v2f a, b;
      a.x = Wq[(size_t)(k + 0) * E_ + rt * 16 + lo];  // A[e, f] = Wq[f, e]
      a.y = Wq[(size_t)(k + 1) * E_ + rt * 16 + lo];
      b.x = Wk[(size_t)(k + 0) * E_ + ct * 16 + lo];  // B[f, e'] = Wk[f, e']
      b.y = Wk[(size_t)(k + 1) * E_ + ct * 16 + lo];
      acc = wmma_f32_4(a, b, acc);
    }
    for (int i = 0; i < 8; ++i)
      Mo[(size_t)(rt * 16 + i + 8 * hi) * E_ + ct * 16 + lo] = acc[i];
  }
}

// -------- Kernel 4: Y[s,v,h] = emb_ln[v] (256x128) @ M[s,h] (128x128) -------
__global__ void __launch_bounds__(256) k_y(const float* __restrict__ embln,
                                           const float* __restrict__ M,
                                           float* __restrict__ Y) {
  int b = blockIdx.x;  // 400 = 2*25*8, b == (s*25+v)*8+h
  int s = b / 200;
  int rem = b % 200;
  int v = rem >> 3, h = rem & 7;
  const float* A = embln + (size_t)v * N_ * E_;
  const float* Mm = M + (size_t)(s * 8 + h) * E_ * E_;
  float* Yo = Y + (size_t)b * N_ * E_;
  int wave = threadIdx.x >> 5, lane = threadIdx.x & 31;
  int lo = lane & 15, hi = lane >> 4;
  for (int j = 0; j < 16; ++j) {
    int t = wave * 16 + j;
    int rt = t >> 3, ct = t & 7;  // 16 row-tiles x 8 col-tiles
    v8f acc = {};
    const float* Arow = A + (size_t)(rt * 16 + lo) * E_;
    for (int k0 = 0; k0 < E_; k0 += 4) {
      int k = k0 + 2 * hi;
      v2f a, b2;
      a.x = Arow[k];
      a.y = Arow[k + 1];
      b2.x = Mm[(size_t)(k + 0) * E_ + ct * 16 + lo];
      b2.y = Mm[(size_t)(k + 1) * E_ + ct * 16 + lo];
      acc = wmma_f32_4(a, b2, acc);
    }
    for (int i = 0; i < 8; ++i)
      Yo[(size_t)(rt * 16 + i + 8 * hi) * E_ + ct * 16 + lo] = acc[i];
  }
}

// --- Kernel 5: S = Y @ ew^T (256x256), column softmax over n, BR weighted ---
// B-operand tile (64 rows x 128 f32 = 32KB, contiguous) is staged into LDS
// once per m-tile via the Tensor Data Mover (TDM) when available.
__global__ void __launch_bounds__(256) k_attn_br(
    const float* __restrict__ Y, const float* __restrict__ embln,
    const int* __restrict__ idxT, const int* __restrict__ idxI,
    const float* __restrict__ bst, const float* __restrict__ bsi,
    float* __restrict__ BRt, float* __restrict__ BRi) {
  extern __shared__ float Slds[];   // 256 rows x 64 cols (64KB dynamic)
  __shared__ float ewsh[64 * E_];   // B-operand tile (32KB)
  __shared__ float wsh[N_];
  __shared__ float redm[4][64], reds[4][64], redw[4][64];
  int bx = blockIdx.x;  // (25+64)*8
  int i = bx >> 3, h = bx & 7;
  int s, v;
  const float* wrow;
  float* brout;
  if (i < R_) {
    s = 0; v = idxT[i * C_];           // index_target_idx[l, 0]
    wrow = bst + (size_t)i * N_;
    brout = BRt + (size_t)i * N_ * H_;
  } else {
    int c = i - R_;
    s = 1; v = idxI[c];                // index_infected_idx[0, c]
    wrow = bsi + (size_t)c * N_;
    brout = BRi + (size_t)c * N_ * H_;
  }
  const float* Yp = Y + (size_t)((s * R_ + v) * H_ + h) * N_ * E_;
  const float* Ew = embln + (size_t)v * N_ * E_;
  int tid = threadIdx.x;
  wsh[tid] = wrow[tid];
  __syncthreads();
  int wave = tid >> 5, lane = tid & 31, lo = lane & 15, hi = lane >> 4;
  const float scale = 0.0883883476483184f;  // 1/sqrt(128)
  int col = tid & 63, q = tid >> 6;
  for (int mt = 0; mt < 4; ++mt) {
    const float* tileSrc = Ew + (size_t)(mt * 64) * E_;
#if USE_TDM
    {
      // Tensor DMA descriptor (D#), CDNA5 ISA:
      // 2D tile: 128 elems x 64 rows, 4-byte elements, row stride 128 elems.
      unsigned lds_off = (unsigned)(uintptr_t)(&ewsh[0]);
      unsigned long long ga = (unsigned long long)(uintptr_t)tileSrc;
      u32x4 g0;
      g0[0] = 1u;                                    // count=1, user mode
      g0[1] = lds_off;                               // lds_addr (bytes)
      g0[2] = (unsigned)(ga & 0xffffffffu);          // global_addr[31:0]
      g0[3] = (unsigned)((ga >> 32) & 0x01ffffffu)   // global_addr[56:32]
              | (2u << 30);                          // type=2 ("image")
      i32x8 g1;
      g1[0] = (int)(2u << 16);    // workgroup_mask=0, data_size=2 (4B)
      g1[1] = (int)(128u << 16);  // tensor_dim0 = 128 (bits 79:48 low part)
      g1[2] = (int)(64u << 16);   // tensor_dim0 hi=0 | tensor_dim1 = 64
      g1[3] = (int)(128u << 16);  // tensor_dim1 hi=0 | tile_dim0 = 128
      g1[4] = 64;                 // tile_dim1 = 64, tile_dim2 = 0
      g1[5] = 128;                // tensor_dim0_stride = 128 (elems)
      g1[6] = 0;
      g1[7] = 0;
      i32x4 gz = {0, 0, 0, 0};
#if __clang_major__ >= 23
      i32x8 gz8 = {0, 0, 0, 0, 0, 0, 0, 0};
      __builtin_amdgcn_tensor_load_to_lds(g0, g1, gz, gz, gz8, 0);
#else
      __builtin_amdgcn_tensor_load_to_lds(g0, g1, gz, gz, 0);
#endif
      __builtin_amdgcn_s_wait_tensorcnt(0);
    }
    __syncthreads();
#else
    for (int jj = tid; jj < 64 * E_; jj += 256) ewsh[jj] = tileSrc[jj];
    __syncthreads();
#endif
    for (int j = 0; j < 8; ++j) {
      int t = wave * 8 + j;
      int rt = t >> 2, ct = t & 3;  // 16 row-tiles x 4 col-tiles
      v8f acc = {};
      const float* Arow = Yp + (size_t)(rt * 16 + lo) * E_;
      const float* Brow = &ewsh[(ct * 16 + lo) * E_];
      for (int k0 = 0; k0 < E_; k0 += 4) {
        int k = k0 + 2 * hi;
        v2f a, b;
        a.x = Arow[k];
        a.y = Arow[k + 1];
        b.x = Brow[k];
        b.y = Brow[k + 1];
        acc = wmma_f32_4(a, b, acc);
      }
      for (int ii = 0; ii < 8; ++ii)
        Slds[(size_t)(rt * 16 + ii + 8 * hi) * 64 + ct * 16 + lo] =
            acc[ii] * scale;
    }
    __syncthreads();
    // column-wise softmax over n (axis=2 in reference) + bst-weighted sum
    float mx = -3.4e38f;
    for (int r = 0; r < 64; ++r) mx = fmaxf(mx, Slds[(q * 64 + r) * 64 + col]);
    redm[q][col] = mx;
    __syncthreads();
    mx = fmaxf(fmaxf(redm[0][col], redm[1][col]),
               fmaxf(redm[2][col], redm[3][col]));
    float se = 0.f, wsum = 0.f;
    for (int r = 0; r < 64; ++r) {
      int n = q * 64 + r;
      float e = expf(Slds[n * 64 + col] - mx);
      se += e;
      wsum += wsh[n] * e;
    }
    reds[q][col] = se;
    redw[q][col] = wsum;
    __syncthreads();
    if (q == 0) {
      float st = reds[0][col] + reds[1][col] + reds[2][col] + reds[3][col];
      float wt = redw[0][col] + redw[1][col] + redw[2][col] + redw[3][col];
      brout[(size_t)(mt * 64 + col) * H_ + h] = wt / st;
    }
    __syncthreads();
  }
}

// ----- Kernel 6a: proj[c] = emb[idx_i[c]*N + biz[c]] @ W_os^T + b_os --------
__global__ void k_proj(const float* __restrict__ emb, const int* __restrict__ idxI,
                       const int* __restrict__ biz, const float* __restrict__ Wos,
                       const float* __restrict__ bos, float* __restrict__ proj) {
  int gid = blockIdx.x * blockDim.x + threadIdx.x;  // 8192
  int c = gid >> 7, f = gid & 127;
  const float* row = emb + ((size_t)idxI[c] * N_ + biz[c]) * E_;
  const float* w = Wos + (size_t)f * E_;
  float acc = bos[f];
  for (int e = 0; e < E_; ++e) acc += row[e] * w[e];
  proj[gid] = acc;
}

// ------- Kernel 6b: BS (cosine), CS (sym-KL), OS (bilinear) pre-LN ----------
__global__ void k_scores(const float* __restrict__ BRt,
                         const float* __restrict__ BRi,
                         const float* __restrict__ cstls,
                         const float* __restrict__ csils,
                         const float* __restrict__ proj,
                         const float* __restrict__ emb,
                         float* __restrict__ out) {
  __shared__ float pr[E_];
  int b = blockIdx.x;  // 25*64
  int l = b >> 6, c = b & 63;
  int n = threadIdx.x;
  if (n < E_) pr[n] = proj[(size_t)c * E_ + n];
  __syncthreads();
  // BS: cosine similarity over h
  const float* a = BRt + ((size_t)l * N_ + n) * H_;
  const float* bb = BRi + ((size_t)c * N_ + n) * H_;
  float dot = 0.f, na = 0.f, nb = 0.f;
  for (int hh = 0; hh < H_; ++hh) {
    float x = a[hh], y = bb[hh];
    dot += x * y; na += x * x; nb += y * y;
  }
  float bsv = dot / fmaxf(sqrtf(na) * sqrtf(nb), 1e-15f);
  // CS: -(KL(m||p)+KL(m||q))/2 with log_m = log_softmax((lp+lq)/2)
  const float* lp = cstls + ((size_t)l * N_ + n) * NC_;
  const float* lq = csils + ((size_t)c * N_ + n) * NC_;
  float mid[NC_];
  float mx = -3.4e38f;
  for (int k = 0; k < NC_; ++k) {
    float m = 0.5f * (lp[k] + lq[k]);
    mid[k] = m;
    mx = fmaxf(mx, m);
  }
  float se = 0.f;
  for (int k = 0; k < NC_; ++k) se += expf(mid[k] - mx);
  float lse = logf(se);
  float klp = 0.f, klq = 0.f;
  for (int k = 0; k < NC_; ++k) {
    float lm = mid[k] - mx - lse;
    float em = expf(lm);
    klp += em * (lm - lp[k]);
    klq += em * (lm - lq[k]);
  }
  float csv = -0.5f * (klp / (float)NC_ + klq / (float)NC_);
  // OS: proj[c] . emb_table[l*256+n]
  const float* er = emb + ((size_t)l * N_ + n) * E_;
  float osv = 0.f;
  for (int e = 0; e < E_; ++e) osv += pr[e] * er[e];
  size_t o = ((size_t)l * C_ + c) * N_ + n;
  const size_t MAT = (size_t)R_ * C_ * N_;
  out[o] = bsv;
  out[MAT + o] = csv;
  out[2 * MAT + o] = osv;
}

// -------- Kernel 7: per-r layernorm over (C,N) for BS, CS, OS (in-place) ----
__global__ void k_final_ln(float* __restrict__ out, const float* __restrict__ w0,
                           const float* __restrict__ b0,
                           const float* __restrict__ w1,
                           const float* __restrict__ b1,
                           const float* __restrict__ w2,
                           const float* __restrict__ b2) {
  __shared__ float rs[256], rq[256];
  int b = blockIdx.x;  // 3*25
  int m = b / R_, r = b % R_;
  const float* w = (m == 0) ? w0 : ((m == 1) ? w1 : w2);
  const float* bi = (m == 0) ? b0 : ((m == 1) ? b1 : b2);
  float* base = out + (size_t)m * R_ * C_ * N_ + (size_t)r * C_ * N_;
  int tid = threadIdx.x;
  const int TOT = C_ * N_;  // 16384
  float s = 0.f, sq = 0.f;
  for (int j = tid; j < TOT; j += 256) {
    float x = base[j];
    s += x;
    sq += x * x;
  }
  rs[tid] = s;
  rq[tid] = sq;
  __syncthreads();
  for (int st = 128; st > 0; st >>= 1) {
    if (tid < st) { rs[tid] += rs[tid + st]; rq[tid] += rq[tid + st]; }
    __syncthreads();
  }
  float mean = rs[0] / (float)TOT;
  float var = rq[0] / (float)TOT - mean * mean;
  float rstd = rsqrtf(var + 1e-5f);
  for (int j = tid; j < TOT; j += 256)
    base[j] = (base[j] - mean) * rstd * w[j] + bi[j];
}

extern "C" void kernel_launch(void* const* d_in, const int* in_sizes, int n_in,
                              void* d_out, int out_size, void* d_ws,
                              size_t ws_size, hipStream_t stream) {
  (void)in_sizes; (void)n_in; (void)out_size; (void)ws_size;
  const float* bs_t = (const float*)d_in[0];
  const float* bs_i = (const float*)d_in[1];
  const float* cs_t = (const float*)d_in[2];
  const float* cs_i = (const float*)d_in[3];
  const int* idxT = (const int*)d_in[4];
  const int* idxI = (const int*)d_in[5];
  const int* biz = (const int*)d_in[6];
  const float* emb = (const float*)d_in[7];
  const float* lnw = (const float*)d_in[8];
  const float* lnb = (const float*)d_in[9];
  const float* WQt = (const float*)d_in[10];
  const float* WKt = (const float*)d_in[11];
  const float* WQi = (const float*)d_in[12];
  const float* WKi = (const float*)d_in[13];
  const float* Wos = (const float*)d_in[14];
  const float* bos = (const float*)d_in[15];
  const float* bsw = (const float*)d_in[16];
  const float* bsb = (const float*)d_in[17];
  const float* csw = (const float*)d_in[18];
  const float* csb = (const float*)d_in[19];
  const float* osw = (const float*)d_in[20];
  const float* osb = (const float*)d_in[21];
  float* out = (float*)d_out;

  float* ws = (float*)d_ws;
  size_t off = 0;
  float* embln = ws + off; off += (size_t)R_ * N_ * E_;           // 819200
  float* Mw = ws + off;    off += (size_t)16 * E_ * E_;           // 262144
  float* Y = ws + off;     off += (size_t)2 * R_ * H_ * N_ * E_;  // 13107200
  float* bst = ws + off;   off += (size_t)R_ * N_;
  float* bsi = ws + off;   off += (size_t)C_ * N_;
  float* cstls = ws + off; off += (size_t)R_ * N_ * NC_;
  float* csils = ws + off; off += (size_t)C_ * N_ * NC_;
  float* BRt = ws + off;   off += (size_t)R_ * N_ * H_;
  float* BRi = ws + off;   off += (size_t)C_ * N_ * H_;
  float* proj = ws + off;  off += (size_t)C_ * E_;
  // total ~15.0M floats (~60 MB) of workspace

  k_emb_ln<<<R_ * N_, E_, 0, stream>>>(emb, lnw, lnb, embln);
  k_bs_mean<<<R_, N_, 0, stream>>>(bs_t, bst, C_ * N_ * 4);
  k_bs_mean<<<C_, N_, 0, stream>>>(bs_i, bsi, N_ * 4);
  k_cs_logsoftmax<<<R_, N_, 0, stream>>>(cs_t, cstls, C_ * N_ * NC_ * 4);
  k_cs_logsoftmax<<<C_, N_, 0, stream>>>(cs_i, csils, N_ * NC_ * 4);
  k_wqk<<<16, 256, 0, stream>>>(WQt, WKt, WQi, WKi, Mw);
  k_y<<<400, 256, 0, stream>>>(embln, Mw, Y);
  k_attn_br<<<(R_ + C_) * H_, 256, (size_t)N_ * 64 * sizeof(float), stream>>>(
      Y, embln, idxT, idxI, bst, bsi, BRt, BRi);
  k_proj<<<32, 256, 0, stream>>>(emb, idxI, biz, Wos, bos, proj);
  k_scores<<<R_ * C_, N_, 0, stream>>>(BRt, BRi, cstls, csils, proj, emb, out);
  k_final_ln<<<3 * R_, 256, 0, stream>>>(out, bsw, bsb, csw, csb, osw, osb);
}